// GINPaper_10737418240832
// MI455X (gfx1250) — compile-verified
//
#include <hip/hip_runtime.h>

#define N_NODES 100000
#define N_EDGES 1600000
#define DIMF    64
#define HID     64
#define NLAYER  3
#define NGRAPH  128
#define NOUT    10
#define BN_EPS  1e-5f

typedef float v2f __attribute__((ext_vector_type(2)));
typedef float v4f __attribute__((ext_vector_type(4)));
typedef float v8f __attribute__((ext_vector_type(8)));

// ---------------------------------------------------------------- copy x -> agg
__global__ __launch_bounds__(256) void copy_x_kernel(const float* __restrict__ x,
                                                     float* __restrict__ agg) {
    int i = blockIdx.x * blockDim.x + threadIdx.x;   // indexes float4; grid exact
    ((v4f*)agg)[i] = ((const v4f*)x)[i];
}

// ------------------------------------------- edge scatter: agg[dst] += x[src]
__global__ __launch_bounds__(256) void scatter_kernel(const float* __restrict__ x,
                                                      const int* __restrict__ ei,
                                                      float* __restrict__ agg) {
    int t = blockIdx.x * blockDim.x + threadIdx.x;   // E*16 threads, grid exact
    int e = t >> 4;
    int q = (t & 15) << 2;                           // float4 chunk within 64 feats
    int s = ei[e];
    int d = ei[N_EDGES + e];
    v4f val = *(const v4f*)(x + (size_t)s * DIMF + q);
    float* out = agg + (size_t)d * DIMF + q;
    atomicAdd(out + 0, val.x);
    atomicAdd(out + 1, val.y);
    atomicAdd(out + 2, val.z);
    atomicAdd(out + 3, val.w);
}

// --------------------------- GEMM1: H = agg @ W1 + b1, accumulate column stats
// block = 128 thr = 4 waves; block handles rows [16*bid,16*bid+16), wave w -> cols [16w,16w+16)
__global__ __launch_bounds__(128) void gemm1_kernel(const float* __restrict__ A,
                                                    const float* __restrict__ W,
                                                    const float* __restrict__ bias,
                                                    float* __restrict__ Hb,
                                                    float* __restrict__ sums,
                                                    float* __restrict__ sumsq) {
    const int wave = threadIdx.x >> 5;
    const int lane = threadIdx.x & 31;
    const int half = lane >> 4;           // 0: K pair {0,1}, 1: K pair {2,3}
    const int l16  = lane & 15;
    const int row0 = blockIdx.x * 16;
    const int n    = wave * 16 + l16;     // output column
    const int arow = row0 + l16;          // A row owned by this lane
    const int kk   = half * 2;

    v8f c = {};
    #pragma unroll
    for (int ks = 0; ks < 16; ++ks) {
        const int k0 = ks * 4 + kk;
        v2f a = *(const v2f*)(A + (size_t)arow * DIMF + k0);
        v2f b;
        b.x = W[(k0 + 0) * HID + n];
        b.y = W[(k0 + 1) * HID + n];
        c = __builtin_amdgcn_wmma_f32_16x16x4_f32(false, a, false, b,
                                                  (short)0, c, false, false);
    }
    const float bn = bias[n];
    const int rbase = row0 + half * 8;    // VGPR r: lanes0-15 row r, lanes16-31 row r+8
    float s = 0.f, sq = 0.f;
    #pragma unroll
    for (int r = 0; r < 8; ++r) {
        float v = c[r] + bn;
        Hb[(size_t)(rbase + r) * HID + n] = v;
        s += v;
        sq += v * v;
    }
    atomicAdd(&sums[n], s);
    atomicAdd(&sumsq[n], sq);
}

// -------------------------------------------------- BN stats -> scale / shift
__global__ void bn_finalize_kernel(const float* __restrict__ sums,
                                   const float* __restrict__ sumsq,
                                   const float* __restrict__ gamma,
                                   const float* __restrict__ beta,
                                   float* __restrict__ scale,
                                   float* __restrict__ shift) {
    int n = threadIdx.x;
    float inv = 1.f / (float)N_NODES;
    float mu  = sums[n] * inv;
    float var = sumsq[n] * inv - mu * mu;
    float sc  = gamma[n] * rsqrtf(var + BN_EPS);
    scale[n] = sc;
    shift[n] = beta[n] - mu * sc;
}

// ----- GEMM2: Xn = relu(BN(H)) @ W2 + b2, fused segment-sum pooling into `pooled`
__global__ __launch_bounds__(128) void gemm2_kernel(const float* __restrict__ Hb,
                                                    const float* __restrict__ scale,
                                                    const float* __restrict__ shift,
                                                    const float* __restrict__ W2,
                                                    const float* __restrict__ bias2,
                                                    const int* __restrict__ batch,
                                                    float* __restrict__ Xn,
                                                    float* __restrict__ pooled) {
    const int wave = threadIdx.x >> 5;
    const int lane = threadIdx.x & 31;
    const int half = lane >> 4;
    const int l16  = lane & 15;
    const int row0 = blockIdx.x * 16;
    const int n    = wave * 16 + l16;
    const int arow = row0 + l16;
    const int kk   = half * 2;

    v8f c = {};
    #pragma unroll
    for (int ks = 0; ks < 16; ++ks) {
        const int k0 = ks * 4 + kk;
        v2f h = *(const v2f*)(Hb + (size_t)arow * HID + k0);
        v2f a;
        a.x = fmaxf(fmaf(h.x, scale[k0 + 0], shift[k0 + 0]), 0.f);  // BN + ReLU on A load
        a.y = fmaxf(fmaf(h.y, scale[k0 + 1], shift[k0 + 1]), 0.f);
        v2f b;
        b.x = W2[(k0 + 0) * HID + n];
        b.y = W2[(k0 + 1) * HID + n];
        c = __builtin_amdgcn_wmma_f32_16x16x4_f32(false, a, false, b,
                                                  (short)0, c, false, false);
    }
    const float bn = bias2[n];
    const int rbase = row0 + half * 8;
    // run-length compressed pooling over sorted batch ids
    int cur = batch[rbase];
    float acc = 0.f;
    #pragma unroll
    for (int r = 0; r < 8; ++r) {
        float v = c[r] + bn;
        Xn[(size_t)(rbase + r) * HID + n] = v;
        int bid = batch[rbase + r];
        if (bid != cur) {
            atomicAdd(&pooled[cur * HID + n], acc);
            acc = 0.f;
            cur = bid;
        }
        acc += v;
    }
    atomicAdd(&pooled[cur * HID + n], acc);
}

// ------------------------------------- score += pooled @ Wout + bout (tiny GEMM)
__global__ __launch_bounds__(256) void score_kernel(const float* __restrict__ pooled,
                                                    const float* __restrict__ Wout,
                                                    const float* __restrict__ bout,
                                                    float* __restrict__ out) {
    int t = blockIdx.x * blockDim.x + threadIdx.x;
    if (t >= NGRAPH * NOUT) return;
    int g = t / NOUT, o = t % NOUT;
    float acc = bout[o];
    #pragma unroll
    for (int k = 0; k < HID; ++k)
        acc = fmaf(pooled[g * HID + k], Wout[k * NOUT + o], acc);
    out[t] += acc;
}

extern "C" void kernel_launch(void* const* d_in, const int* in_sizes, int n_in,
                              void* d_out, int out_size, void* d_ws, size_t ws_size,
                              hipStream_t stream) {
    const float* x     = (const float*)d_in[0];
    const int*   ei    = (const int*)d_in[1];
    const int*   batch = (const int*)d_in[2];
    const float* W1    = (const float*)d_in[3];
    const float* b1    = (const float*)d_in[4];
    const float* gamma = (const float*)d_in[5];
    const float* beta  = (const float*)d_in[6];
    const float* W2    = (const float*)d_in[7];
    const float* b2    = (const float*)d_in[8];
    const float* Wout  = (const float*)d_in[9];
    const float* bout  = (const float*)d_in[10];
    float* out = (float*)d_out;

    float* ws     = (float*)d_ws;
    const size_t NF = (size_t)N_NODES * HID;
    float* agg    = ws;
    float* hb     = ws + NF;
    float* xn     = ws + 2 * NF;
    float* sums   = ws + 3 * NF;           // 64
    float* sumsq  = sums + HID;            // 64
    float* scale  = sums + 2 * HID;        // 64
    float* shift  = sums + 3 * HID;        // 64
    float* pooled = sums + 4 * HID;        // G*64

    hipMemsetAsync(out, 0, NGRAPH * NOUT * sizeof(float), stream);

    const float* xcur = x;
    for (int l = 0; l < NLAYER; ++l) {
        hipMemsetAsync(sums, 0, (4 * HID + NGRAPH * HID) * sizeof(float), stream);
        copy_x_kernel<<<(N_NODES * DIMF / 4) / 256, 256, 0, stream>>>(xcur, agg);
        scatter_kernel<<<(N_EDGES * 16) / 256, 256, 0, stream>>>(xcur, ei, agg);
        gemm1_kernel<<<N_NODES / 16, 128, 0, stream>>>(
            agg, W1 + l * DIMF * HID, b1 + l * HID, hb, sums, sumsq);
        bn_finalize_kernel<<<1, HID, 0, stream>>>(sums, sumsq,
            gamma + l * HID, beta + l * HID, scale, shift);
        gemm2_kernel<<<N_NODES / 16, 128, 0, stream>>>(
            hb, scale, shift, W2 + l * HID * HID, b2 + l * HID, batch, xn, pooled);
        score_kernel<<<(NGRAPH * NOUT + 255) / 256, 256, 0, stream>>>(
            pooled, Wout + l * HID * NOUT, bout + l * NOUT, out);
        xcur = xn;
    }
}